// RANSAC_66675072303601
// MI455X (gfx1250) — compile-verified
//
#include <hip/hip_runtime.h>

typedef float v2f __attribute__((ext_vector_type(2)));
typedef float v8f __attribute__((ext_vector_type(8)));

#define PATCH 14.0f
#define THR2 25.0f            // PIXEL_THRESHOLD^2
#define BATCH 16
#define NPTS 1024
#define NT 256                // threads per block (8 waves)
#define NW 8
#define MT_PER_WAVE 8         // 64 row-tiles / 8 waves

__global__ __launch_bounds__(NT)
void ransac_wmma_kernel(const float* __restrict__ src,   // (B,N,2)
                        const float* __restrict__ tar,   // (B,N,2)
                        const float* __restrict__ scl,   // (B,N)
                        const float* __restrict__ cs,    // (B,N,2)  cos,sin
                        const float* __restrict__ sco,   // (B,N)
                        float* __restrict__ out)
{
    __shared__ float sX[NPTS], sY[NPTS], tX[NPTS], tY[NPTS], scS[NPTS];
    __shared__ float A00[NPTS], A10[NPTS], TXs[NPTS], TYs[NPTS];
    __shared__ float scoreInl[NPTS];
    __shared__ float redVal[NT];
    __shared__ int   redIdx[NT];
    __shared__ int   cnt[NT];
    __shared__ int   bestIdxSh;

    const int b   = blockIdx.x;
    const int tid = threadIdx.x;

    const float* srcB = src + b * NPTS * 2;
    const float* tarB = tar + b * NPTS * 2;
    const float* sclB = scl + b * NPTS;
    const float* csB  = cs  + b * NPTS * 2;
    const float* scoB = sco + b * NPTS;

    // ---------------- Phase 1: stage points + per-hypothesis transform params
#pragma unroll
    for (int j = 0; j < NPTS / NT; ++j) {
        int n = j * NT + tid;
        float sx = srcB[2*n+0] * PATCH, sy = srcB[2*n+1] * PATCH;
        float px = tarB[2*n+0] * PATCH, py = tarB[2*n+1] * PATCH;
        float sc = sclB[n];
        float co = csB[2*n+0], si = csB[2*n+1];
        float a00 = sc * co;      // A = [[a00, -a10],[a10, a00]]
        float a10 = sc * si;
        float trx = px - (a00 * sx - a10 * sy);
        float trY = py - (a10 * sx + a00 * sy);
        sX[n] = sx;  sY[n] = sy;  tX[n] = px;  tY[n] = py;
        scS[n] = scoB[n];
        A00[n] = a00; A10[n] = a10; TXs[n] = trx; TYs[n] = trY;
    }
    __syncthreads();

    // ---------------- Phase 2: score all hypotheses via V_WMMA_F32_16X16X4_F32
    const int lane = tid & 31;
    const int wave = tid >> 5;
    const int li   = lane & 15;
    const bool hi  = lane >= 16;

    for (int mi = 0; mi < MT_PER_WAVE; ++mi) {
        const int m0 = (wave * MT_PER_WAVE + mi) * 16;
        const int m  = m0 + li;

        // A rows (K=0..3): ex -> [a00, -a10, tx, -1] ; ey -> [a10, a00, ty, -1]
        // lanes 0-15 hold K=0 (v0), K=1 (v1); lanes 16-31 hold K=2 (v0), K=3 (v1)
        v2f aex, aey;
        aex.x = hi ? TXs[m] :  A00[m];
        aex.y = hi ? -1.0f  : -A10[m];
        aey.x = hi ? TYs[m] :  A10[m];
        aey.y = hi ? -1.0f  :  A00[m];

        float acc[8];
#pragma unroll
        for (int r = 0; r < 8; ++r) acc[r] = 0.0f;

        for (int k0 = 0; k0 < NPTS; k0 += 16) {
            const int col = k0 + li;
            float csx = sX[col], csy = sY[col];
            float ctx = tX[col], cty = tY[col];
            float csc = scS[col];

            // B cols (K=0..3): ex -> [sx, sy, 1, tkx] ; ey -> [sx, sy, 1, tky]
            v2f bex, bey;
            bex.x = hi ? 1.0f : csx;  bex.y = hi ? ctx : csy;
            bey.x = hi ? 1.0f : csx;  bey.y = hi ? cty : csy;

            v8f zero = {};
            v8f dex = __builtin_amdgcn_wmma_f32_16x16x4_f32(
                false, aex, false, bex, (short)0, zero, false, false);
            v8f dey = __builtin_amdgcn_wmma_f32_16x16x4_f32(
                false, aey, false, bey, (short)0, zero, false, false);

#pragma unroll
            for (int r = 0; r < 8; ++r) {
                float ex = dex[r], ey = dey[r];
                float e2 = ex * ex + ey * ey;
                acc[r] += (e2 <= THR2) ? csc : 0.0f;
            }
        }

        // reduce across the 16 lanes of each half-wave (masks < 16 stay in-group)
#pragma unroll
        for (int r = 0; r < 8; ++r) {
            float v = acc[r];
            v += __shfl_xor(v, 1);
            v += __shfl_xor(v, 2);
            v += __shfl_xor(v, 4);
            v += __shfl_xor(v, 8);
            acc[r] = v;
        }
        if (li == 0) {
#pragma unroll
            for (int r = 0; r < 8; ++r) {
                int m2 = m0 + r + (hi ? 8 : 0);
                scoreInl[m2] = acc[r] - scS[m2];   // remove self-match
            }
        }
    }
    __syncthreads();

    // ---------------- Phase 3: argmax with first-index tie-break
    {
        float bv = -1.0f; int bi = 0;
#pragma unroll
        for (int j = 0; j < NPTS / NT; ++j) {
            int n = tid * (NPTS / NT) + j;       // contiguous -> index-ordered
            float v = scoreInl[n];
            if (v > bv) { bv = v; bi = n; }
        }
        redVal[tid] = bv; redIdx[tid] = bi;
    }
    __syncthreads();
    if (tid == 0) {
        float bv = redVal[0]; int bi = redIdx[0];
        for (int tt = 1; tt < NT; ++tt)
            if (redVal[tt] > bv) { bv = redVal[tt]; bi = redIdx[tt]; }
        bestIdxSh = bi;
        // Ms and failed flag
        float a00 = A00[bi], a10 = A10[bi];
        float* M = out + b * 9;
        M[0] = a00;  M[1] = -a10; M[2] = TXs[bi];
        M[3] = a10;  M[4] = a00;  M[5] = TYs[bi];
        M[6] = 0.0f; M[7] = 0.0f; M[8] = 1.0f;
        out[BATCH * 9 + b] = (bv == 0.0f) ? 1.0f : 0.0f;
    }
    __syncthreads();

    // ---------------- Phase 4: stable inlier compaction
    float* oSrc = out + BATCH * 9 + BATCH + b * NPTS * 2;
    float* oTar = out + BATCH * 9 + BATCH + BATCH * NPTS * 2 + b * NPTS * 2;
    float* oSc  = out + BATCH * 9 + BATCH + 2 * BATCH * NPTS * 2 + b * NPTS;

    // default fill (covers non-inlier slots and the sentinel row)
#pragma unroll
    for (int j = 0; j < NPTS / NT; ++j) {
        int n = j * NT + tid;
        oSrc[2*n] = -1.0f; oSrc[2*n+1] = -1.0f;
        oTar[2*n] = -1.0f; oTar[2*n+1] = -1.0f;
        oSc[n]    =  0.0f;
    }

    const int bi = bestIdxSh;
    const float a00 = A00[bi], a01 = -A10[bi], a10 = A10[bi], a11 = A00[bi];
    const float trx = TXs[bi], trY = TYs[bi];

    int flags = 0, c = 0;
#pragma unroll
    for (int j = 0; j < NPTS / NT; ++j) {
        int k = tid * (NPTS / NT) + j;           // contiguous per-thread range
        float ex = a00 * sX[k] + a01 * sY[k] + trx - tX[k];
        float ey = a10 * sX[k] + a11 * sY[k] + trY - tY[k];
        bool f = (k != bi) && (ex * ex + ey * ey <= THR2);
        flags |= (int)f << j;
        c += (int)f;
    }
    cnt[tid] = c;
    __syncthreads();
    if (tid == 0) {                               // exclusive scan (256 entries)
        int s = 0;
        for (int tt = 0; tt < NT; ++tt) { int cc = cnt[tt]; cnt[tt] = s; s += cc; }
    }
    __syncthreads();

    int base = cnt[tid];
#pragma unroll
    for (int j = 0; j < NPTS / NT; ++j) {
        if (flags & (1 << j)) {
            int k = tid * (NPTS / NT) + j;
            oSrc[2*base]   = srcB[2*k];
            oSrc[2*base+1] = srcB[2*k+1];
            oTar[2*base]   = tarB[2*k];
            oTar[2*base+1] = tarB[2*k+1];
            oSc[base]      = scoB[k];
            ++base;
        }
    }
}

extern "C" void kernel_launch(void* const* d_in, const int* in_sizes, int n_in,
                              void* d_out, int out_size, void* d_ws, size_t ws_size,
                              hipStream_t stream) {
    const float* src = (const float*)d_in[0];   // src_pts   (B,N,2)
    const float* tar = (const float*)d_in[1];   // tar_pts   (B,N,2)
    const float* scl = (const float*)d_in[2];   // relScales (B,N)
    const float* cs  = (const float*)d_in[3];   // relInplanes (B,N,2)
    const float* sco = (const float*)d_in[4];   // scores    (B,N)
    (void)in_sizes; (void)n_in; (void)out_size; (void)d_ws; (void)ws_size;
    ransac_wmma_kernel<<<BATCH, NT, 0, stream>>>(src, tar, scl, cs, sco, (float*)d_out);
}